// CustomODELoss_32074815766941
// MI455X (gfx1250) — compile-verified
//
#include <hip/hip_runtime.h>

// ---------------------------------------------------------------------------
// CustomODELoss for MI455X (gfx1250, wave32).
//
// Term 1 (dominant): mean((pred-target)^2) over 33.5M f32 = 256 MiB streamed
// once -> pure HBM-bound (~11.5us floor at 23.3 TB/s). Strategy:
//   * b128 non-temporal global loads (read-once stream; don't rinse L2)
//   * squared diffs accumulated with V_WMMA_F32_16X16X4_F32, B = ones(4x16):
//       D[m][n] += sum_k A[m][k]   (64-element accumulating add per WMMA)
//   * layout-proof extraction: with B==ones all columns of D are equal, so
//     sum of ALL 256 D entries == 16 * total. Per-lane sum of the 8 C regs,
//     5-step shfl_xor butterfly across the wave32, scale by 1/16.
//   * deterministic 2-stage tree reduction (no float atomics).
// Terms 2/3: one thread per row, O(B) work.
// ---------------------------------------------------------------------------

typedef __attribute__((ext_vector_type(2))) float v2f;
typedef __attribute__((ext_vector_type(4))) float v4f;
typedef __attribute__((ext_vector_type(8))) float v8f;

#ifndef GRID_A
#define GRID_A 1024   // term-1 partials count (fixed -> deterministic)
#endif

__device__ __forceinline__ v8f wmma_acc_f32(v2f a, v2f ones, v8f c) {
    // D = A(16x4) x ones(4x16) + C ; accumulates sum_k A[m][k] into row m.
    return __builtin_amdgcn_wmma_f32_16x16x4_f32(
        /*neg_a=*/false, a, /*neg_b=*/false, ones,
        /*c_mod=*/(short)0, c, /*reuse_a=*/false, /*reuse_b=*/false);
}

__global__ __launch_bounds__(256)
void ode_loss_sse_kernel(const float* __restrict__ pred,
                         const float* __restrict__ targ,
                         float* __restrict__ partials,   // [GRID_A]
                         long long n4, long long ntotal)
{
    const long long tid    = (long long)blockIdx.x * blockDim.x + threadIdx.x;
    const long long stride = (long long)gridDim.x * blockDim.x;

    v8f c0 = {};
    v8f c1 = {};
    v2f ones; ones.x = 1.0f; ones.y = 1.0f;

    const v4f* __restrict__ p4 = (const v4f*)pred;
    const v4f* __restrict__ t4 = (const v4f*)targ;

    // Uniform trip count across the wave (stride is wave-uniform) -> EXEC all
    // ones at every WMMA, as required by the ISA.
    for (long long i = tid; i < n4; i += stride) {
        v4f p = __builtin_nontemporal_load(p4 + i);   // global_load_b128 TH=NT
        v4f t = __builtin_nontemporal_load(t4 + i);
        float d0 = p.x - t.x;
        float d1 = p.y - t.y;
        float d2 = p.z - t.z;
        float d3 = p.w - t.w;
        v2f a0; a0.x = d0 * d0; a0.y = d1 * d1;
        v2f a1; a1.x = d2 * d2; a1.y = d3 * d3;
        c0 = wmma_acc_f32(a0, ones, c0);   // independent accumulator chains
        c1 = wmma_acc_f32(a1, ones, c1);   // keep the XDL pipe busy
    }

    // Scalar tail (ntotal % 4). Divergent loop, but fully reconverged before
    // the WMMA below, which folds the leftovers into the accumulator.
    float extra = 0.0f;
    for (long long j = 4 * n4 + tid; j < ntotal; j += stride) {
        float d = pred[j] - targ[j];
        extra = fmaf(d, d, extra);
    }
    v2f at; at.x = extra; at.y = 0.0f;
    c0 = wmma_acc_f32(at, ones, c0);

    // Extraction: sum of all 256 D entries = 16 * total (columns identical).
    v8f cs = c0 + c1;
    float lane_sum = 0.0f;
#pragma unroll
    for (int k = 0; k < 8; ++k) lane_sum += cs[k];
#pragma unroll
    for (int off = 16; off >= 1; off >>= 1)
        lane_sum += __shfl_xor(lane_sum, off, 32);
    float wave_total = lane_sum * 0.0625f;   // /16

    __shared__ float lred[8];                // 256 threads = 8 wave32
    const int wave = threadIdx.x >> 5;
    if ((threadIdx.x & 31) == 0) lred[wave] = wave_total;
    __syncthreads();
    if (threadIdx.x == 0) {
        float s = 0.0f;
#pragma unroll
        for (int w = 0; w < 8; ++w) s += lred[w];
        partials[blockIdx.x] = s;
    }
}

__global__ __launch_bounds__(256)
void ode_loss_bc_kernel(const float* __restrict__ pred,
                        const float* __restrict__ c_in,
                        const float* __restrict__ x,
                        float* __restrict__ ws_t2,   // [gridDim.x]
                        float* __restrict__ ws_t3,   // [gridDim.x]
                        int Brows, int N)
{
    const int b = blockIdx.x * blockDim.x + threadIdx.x;
    float t2 = 0.0f, t3 = 0.0f;
    if (b < Brows) {
        const float c  = c_in[b];
        const float dx = (N > 1) ? (x[1] - x[0]) : 1.0f;

        // Analytic nearest-index guess, refined over a +/-2 window against
        // the true grid values; ascending scan with strict '<' replicates
        // argmin's first-hit tie-break.
        int i0 = (int)floorf(c / dx + 0.5f);
        i0 = min(max(i0, 0), N - 1);
        int lo = max(i0 - 2, 0), hi = min(i0 + 2, N - 1);
        int best = lo;
        float bd = fabsf(x[lo] - c);
        for (int j = lo + 1; j <= hi; ++j) {
            float d = fabsf(x[j] - c);
            if (d < bd) { bd = d; best = j; }
        }

        const float* __restrict__ row = pred + (long long)b * N;
        const float fc = row[best];
        const int ip = min(best + 1, N - 1);
        const int im = max(best - 1, 0);
        const float denom = (float)(ip - im) * dx;   // 2*dx interior, dx edge
        const float fpc = (row[ip] - row[im]) / denom;

        const float e2 = fc - 1.0f;                  // Y0_FIXED = 1
        t2 = e2 * e2;
        t3 = fpc * fpc;                              // YP0_FIXED = 0
    }

    __shared__ float s2[256], s3[256];
    s2[threadIdx.x] = t2;
    s3[threadIdx.x] = t3;
    __syncthreads();
#pragma unroll
    for (int off = 128; off >= 1; off >>= 1) {
        if (threadIdx.x < off) {
            s2[threadIdx.x] += s2[threadIdx.x + off];
            s3[threadIdx.x] += s3[threadIdx.x + off];
        }
        __syncthreads();
    }
    if (threadIdx.x == 0) {
        ws_t2[blockIdx.x] = s2[0];
        ws_t3[blockIdx.x] = s3[0];
    }
}

__global__ __launch_bounds__(256)
void ode_loss_final_kernel(const float* __restrict__ ws,
                           float* __restrict__ out,
                           int nb1, int nb2, long long ntotal, int Brows)
{
    __shared__ float r1[256], r2[256], r3[256];
    const int t = threadIdx.x;
    float s1 = 0.0f, s2 = 0.0f, s3 = 0.0f;
    for (int i = t; i < nb1; i += 256) s1 += ws[i];
    for (int i = t; i < nb2; i += 256) {
        s2 += ws[nb1 + i];
        s3 += ws[nb1 + nb2 + i];
    }
    r1[t] = s1; r2[t] = s2; r3[t] = s3;
    __syncthreads();
#pragma unroll
    for (int off = 128; off >= 1; off >>= 1) {
        if (t < off) {
            r1[t] += r1[t + off];
            r2[t] += r2[t + off];
            r3[t] += r3[t + off];
        }
        __syncthreads();
    }
    if (t == 0) {
        // ALPHA = BETA = 1
        out[0] = r1[0] / (float)ntotal
               + r2[0] / (float)Brows
               + r3[0] / (float)Brows;
    }
}

extern "C" void kernel_launch(void* const* d_in, const int* in_sizes, int n_in,
                              void* d_out, int out_size, void* d_ws, size_t ws_size,
                              hipStream_t stream)
{
    const float* pred = (const float*)d_in[0];   // [B, N]
    const float* targ = (const float*)d_in[1];   // [B, N]
    const float* c_in = (const float*)d_in[2];   // [B]
    const float* x    = (const float*)d_in[3];   // [N]

    const long long ntotal = (long long)in_sizes[0];
    const int Brows        = in_sizes[2];
    const int N            = in_sizes[3];
    const long long n4     = ntotal >> 2;

    float* ws = (float*)d_ws;
    const int gridB = (Brows + 255) / 256;

    // ws layout: [0, GRID_A) term1 | [GRID_A, GRID_A+gridB) t2 | next gridB t3
    ode_loss_sse_kernel<<<GRID_A, 256, 0, stream>>>(pred, targ, ws, n4, ntotal);
    ode_loss_bc_kernel<<<gridB, 256, 0, stream>>>(pred, c_in, x,
                                                  ws + GRID_A, ws + GRID_A + gridB,
                                                  Brows, N);
    ode_loss_final_kernel<<<1, 256, 0, stream>>>(ws, (float*)d_out,
                                                 GRID_A, gridB, ntotal, Brows);
}